// Capsule_13692355740297
// MI455X (gfx1250) — compile-verified
//
#include <hip/hip_runtime.h>
#include <cstdint>

// CapsNet routing for MI455X (gfx1250, wave32).
// - FP32 WMMA GEMM (v_wmma_f32_16x16x4_f32) materializes u_hat[B,N,I,D]
//   (160MB -> L2-resident on a 192MB-L2 part).
// - u_vecs streamed once from HBM with non-temporal async-to-LDS loads.
// - Three fused routing passes stage u_hat chunks into LDS via
//   global_load_async_to_lds_b128 (ASYNCcnt), then do logits+softmax and the
//   weighted sum entirely from LDS (halves L2 traffic vs re-reading).

typedef float v2f __attribute__((ext_vector_type(2)));
typedef float v4f __attribute__((ext_vector_type(4)));
typedef float v8f __attribute__((ext_vector_type(8)));
typedef int   v4i __attribute__((ext_vector_type(4)));

#define AS3 __attribute__((address_space(3)))

#if __has_builtin(__builtin_amdgcn_global_load_async_to_lds_b128)
#define HAVE_ASYNC_LDS 1
#else
#warning "gfx1250 async-to-LDS builtin not found; falling back to VGPR staging"
#define HAVE_ASYNC_LDS 0
#endif

__device__ __forceinline__ void async_wait_all() {
  asm volatile("s_wait_asynccnt 0x0" ::: "memory");
}

#define B_SZ 64
#define I_SZ 4096
#define DIN  128
#define NCAP 10
#define DCAP 16
#define NDK  (NCAP * DCAP)       // 160
#define ICH  128                 // i-chunk per routing block
#define CHUNKS (I_SZ / ICH)      // 32
#define ISTRIDE 20               // LDS i-stride (words): 16B-aligned, low-conflict
#define NSTRIDE (ICH * ISTRIDE + 16)  // per-capsule stride; +16 words => lane
                                      // halves (n, n+1) hit disjoint banks
#define EPS_F 1e-7f

// ---------------------------------------------------------------------------
// GEMM: u_hat[b][n][i][d] = sum_k u_vecs[b][i][k] * W[k][n*16+d]
// 2048 blocks x 256 threads (8 waves); each block covers 128 rows of one b.
// A tile async-staged into 64KB LDS with a +4*row rotation swizzle; each wave
// preloads its 16-row A fragments for all 32 k-steps (64 VGPRs) and runs the
// 10 capsule column tiles with 32 chained V_WMMA_F32_16X16X4_F32 each.
// ---------------------------------------------------------------------------
__global__ __launch_bounds__(256) void caps_gemm(const float* __restrict__ U,
                                                 const float* __restrict__ W,
                                                 float* __restrict__ UH) {
  __shared__ float As[128 * 128];  // (r,c) -> As[r*128 + ((c+4r)&127)]

  const int t    = threadIdx.x;
  const int row0 = blockIdx.x * 128;                 // global row = b*4096 + i

  const v4f* gsrc = (const v4f*)(U + (size_t)row0 * DIN);
  #pragma unroll
  for (int it = 0; it < 16; ++it) {
    int flat = t * 4 + it * 1024;                    // element offset in tile
    int r = flat >> 7;
    int c = flat & 127;
    float* lp = &As[r * 128 + ((c + 4 * r) & 127)];
#if HAVE_ASYNC_LDS
    // cpol=1 (TH=NT): u_vecs is read exactly once; don't pollute L2, where
    // u_hat must stay resident for the routing phase.
    __builtin_amdgcn_global_load_async_to_lds_b128(
        (v4i*)(gsrc + (flat >> 2)), (AS3 v4i*)lp, 0, 1);
#else
    v4f val = __builtin_nontemporal_load(gsrc + (flat >> 2));
    *(v4f*)lp = val;
#endif
  }
#if HAVE_ASYNC_LDS
  async_wait_all();
#endif
  __syncthreads();

  const int w  = t >> 5;       // wave 0..7
  const int l  = t & 31;       // lane
  const int m  = l & 15;
  const int h  = l >> 4;       // lane half selects K phase (ISA f32 layout)
  const int m0 = w * 16;

  // ISA 16x4 f32 A layout: v0 = K = 4s + 2h, v1 = K = 4s + 2h + 1.
  v2f afrag[32];
  {
    const float* arow = &As[(m0 + m) * 128];
    const int rot = 4 * (m0 + m);
    #pragma unroll
    for (int s = 0; s < 32; ++s) {
      int kk = (4 * s + 2 * h + rot) & 127;          // even => pair contiguous
      afrag[s] = *(const v2f*)(arow + kk);
    }
  }

  const int b_idx  = row0 >> 12;                     // /4096
  const int i_base = row0 & (I_SZ - 1);

  for (int n = 0; n < NCAP; ++n) {
    v8f acc = {};
    const float* wcol = W + n * DCAP + m;            // column n*16 + (lane&15)
    #pragma unroll
    for (int s = 0; s < 32; ++s) {
      int kk = 4 * s + 2 * h;
      v2f bfrag;                                     // B 4x16 f32, same K phase
      bfrag.x = wcol[(size_t)kk * NDK];
      bfrag.y = wcol[(size_t)(kk + 1) * NDK];
      acc = __builtin_amdgcn_wmma_f32_16x16x4_f32(
          false, afrag[s], false, bfrag, (short)0, acc, false, false);
    }
    // D layout: N = lane&15 (=d), M = vgpr + 8*(lane>=16); scatter into
    // [B,N,I,D] with 64B-coalesced half-wave stores.
    float* obase = UH +
        (((size_t)b_idx * NCAP + n) * I_SZ + (size_t)(i_base + m0 + 8 * h)) * DCAP + m;
    #pragma unroll
    for (int v = 0; v < 8; ++v) {
      obase[(size_t)v * DCAP] = acc[v];
    }
  }
}

// ---------------------------------------------------------------------------
// Fused routing pass. Grid (chunk 0..31, b 0..63), 256 threads.
// Stage u_hat[b, :, i0:i0+128, :] (80KB) into padded LDS once (async),
// then: pass 1 = thread-per-i logits + softmax over the 10 capsules;
//       pass 2 = thread-per-(n,d) weighted partial sums, all from LDS.
// ---------------------------------------------------------------------------
__global__ __launch_bounds__(256) void caps_route(const float* __restrict__ UH,
                                                  const float* __restrict__ Vin,
                                                  float* __restrict__ Part,
                                                  int first) {
  __shared__ float uh_l[NCAP * NSTRIDE];   // [n][i][d], i-stride 20, n-pad 16
  __shared__ float v_l[NDK];
  __shared__ float c_l[ICH * NCAP];

  const int t     = threadIdx.x;
  const int chunk = blockIdx.x;
  const int b     = blockIdx.y;
  const int i0    = chunk * ICH;

  // ---- async stage: NCAP segments of ICH*16 contiguous floats each
  for (int u = t; u < NCAP * ICH * 4; u += 256) {    // u indexes float4 units
    int n       = u >> 9;                            // / (ICH*4)
    int r       = u & (ICH * 4 - 1);
    int i_local = r >> 2;
    int q       = r & 3;
    const float* g = UH + (((size_t)b * NCAP + n) * I_SZ + i0 + i_local) * DCAP + q * 4;
    float* lp = &uh_l[n * NSTRIDE + i_local * ISTRIDE + q * 4];
#if HAVE_ASYNC_LDS
    __builtin_amdgcn_global_load_async_to_lds_b128(
        (v4i*)g, (AS3 v4i*)lp, 0, 0);
#else
    *(v4f*)lp = *(const v4f*)g;
#endif
  }
  if (!first && t < NDK) v_l[t] = Vin[b * NDK + t];
#if HAVE_ASYNC_LDS
  async_wait_all();
#endif
  __syncthreads();

  // ---- pass 1: per-i coupling coefficients (softmax over capsules)
  if (t < ICH) {
    if (first) {
      #pragma unroll
      for (int n = 0; n < NCAP; ++n) c_l[t * NCAP + n] = 0.1f;   // softmax(0)
    } else {
      float logit[NCAP];
      #pragma unroll
      for (int n = 0; n < NCAP; ++n) {
        const v4f* p  = (const v4f*)&uh_l[n * NSTRIDE + t * ISTRIDE];
        const v4f* vv = (const v4f*)&v_l[n * DCAP];
        float acc = 0.f;
        #pragma unroll
        for (int q = 0; q < 4; ++q) {
          v4f x = p[q], y = vv[q];
          acc += x.x * y.x + x.y * y.y + x.z * y.z + x.w * y.w;
        }
        logit[n] = acc;
      }
      float mx = logit[0];
      #pragma unroll
      for (int n = 1; n < NCAP; ++n) mx = fmaxf(mx, logit[n]);
      float ssum = 0.f;
      #pragma unroll
      for (int n = 0; n < NCAP; ++n) { logit[n] = expf(logit[n] - mx); ssum += logit[n]; }
      float inv = 1.f / ssum;
      #pragma unroll
      for (int n = 0; n < NCAP; ++n) c_l[t * NCAP + n] = logit[n] * inv;
    }
  }
  __syncthreads();

  // ---- pass 2: s_partial[n][d] = sum_i c[i][n] * u_hat[b,n,i,d] (from LDS)
  if (t < NDK) {
    const int n = t >> 4, d = t & 15;
    const float* p = &uh_l[n * NSTRIDE + d];
    float acc = 0.f;
    #pragma unroll 8
    for (int ii = 0; ii < ICH; ++ii)
      acc += c_l[ii * NCAP + n] * p[ii * ISTRIDE];
    Part[((size_t)b * NDK + t) * CHUNKS + chunk] = acc;
  }
}

// ---------------------------------------------------------------------------
// Reduce chunk partials + squash. g = b*160 + n*16 + d; 16-lane groups share
// one (b,n) => norm via 16-wide xor-shuffle reduction (wave32-safe).
// ---------------------------------------------------------------------------
__global__ __launch_bounds__(256) void caps_squash(const float* __restrict__ Part,
                                                   float* __restrict__ Vout) {
  const int g = blockIdx.x * 256 + threadIdx.x;      // 0..10239
  const float* p = Part + (size_t)g * CHUNKS;
  float s = 0.f;
  #pragma unroll
  for (int c = 0; c < CHUNKS; ++c) s += p[c];

  float sq = s * s;
  #pragma unroll
  for (int mask = 1; mask < 16; mask <<= 1) sq += __shfl_xor(sq, mask, 32);

  float sqn   = sq + EPS_F;
  float coeff = sqrtf(sqn) / (1.f + sqn);
  Vout[g] = coeff * s;
}

// ---------------------------------------------------------------------------
extern "C" void kernel_launch(void* const* d_in, const int* in_sizes, int n_in,
                              void* d_out, int out_size, void* d_ws, size_t ws_size,
                              hipStream_t stream) {
  const float* U = (const float*)d_in[0];   // [64,4096,128]
  const float* W = (const float*)d_in[1];   // [128,160]
  float* out = (float*)d_out;               // [64,10,16]

  float* UH   = (float*)d_ws;                               // 64*10*4096*16 f32
  size_t uhN  = (size_t)B_SZ * NCAP * I_SZ * DCAP;
  float* Part = UH + uhN;                                   // 64*160*32 f32
  float* V    = Part + (size_t)B_SZ * NDK * CHUNKS;         // 64*160 f32
  (void)ws_size; (void)in_sizes; (void)n_in; (void)out_size;

  caps_gemm<<<dim3((B_SZ * I_SZ) / 128), 256, 0, stream>>>(U, W, UH);

  for (int it = 0; it < 3; ++it) {
    caps_route<<<dim3(CHUNKS, B_SZ), 256, 0, stream>>>(UH, V, Part, it == 0 ? 1 : 0);
    caps_squash<<<dim3((B_SZ * NDK) / 256), 256, 0, stream>>>(Part, it == 2 ? out : V);
  }
}